// DWA_CNN_29472065585783
// MI455X (gfx1250) — compile-verified
//
#include <hip/hip_runtime.h>

typedef __attribute__((ext_vector_type(2))) float v2f;
typedef __attribute__((ext_vector_type(8))) float v8f;

#define BIGV 1234567891011.0f

static constexpr int B_ = 32;
static constexpr int T_ = 2048;
static constexpr int C_ = 128;
static constexpr int K_ = 3;
static constexpr int F_ = 8;
static constexpr int P_ = T_ - K_ + 1;                     // 2046
static constexpr int NW_ = K_ * F_;                        // 24 gemm columns
static constexpr int TILE_P = 14;                          // windows per 16-row tile
static constexpr int NTILES = (P_ + TILE_P - 1) / TILE_P;  // 147
static constexpr int WAVES = 8;                            // waves per block (256 thr)
static constexpr int GROUPS = (NTILES + WAVES - 1) / WAVES;// 19 blocks per batch
static constexpr int BT_PITCH = 130;                       // even, bank-conflict-free

__global__ __launch_bounds__(256) void dwa_cnn_fused(
    const float* __restrict__ x,     // [B, T, C]
    const float* __restrict__ w,     // [K, C, F]
    const float* __restrict__ bias,  // [F]
    float* __restrict__ out)         // [B, P, F]
{
    // LDS: padded/transposed filter matrix Bt[n][c] (n = j*8+f, zero pad n>=24),
    // filter norms, per-wave G tile (16 rows x 24 cols), per-wave row norms.
    __shared__ __align__(16) float Bt[32 * BT_PITCH];
    __shared__ float wn[32];
    __shared__ float Gl[WAVES][16 * NW_];
    __shared__ float xnl[WAVES][16];

    const int tid  = threadIdx.x;
    const int lane = tid & 31;
    const int wave = tid >> 5;

    const int bb  = blockIdx.x / GROUPS;
    const int grp = blockIdx.x % GROUPS;

    // ---- stage filters into LDS: Bt[n][c] = w[(n/8)*C + c]*F + (n%8), pad with 0
    for (int idx = tid; idx < 32 * C_; idx += 256) {
        const int n = idx >> 7;
        const int c = idx & (C_ - 1);
        float v = 0.0f;
        if (n < NW_) v = w[(((size_t)(n >> 3)) * C_ + c) * F_ + (n & 7)];
        Bt[n * BT_PITCH + c] = v;
    }
    __syncthreads();
    if (tid < NW_) {
        float s = 0.0f;
        #pragma unroll 8
        for (int c = 0; c < C_; ++c) { float v = Bt[tid * BT_PITCH + c]; s += v * v; }
        wn[tid] = s;
    }
    __syncthreads();

    // ---- this wave's 16-row tile of x (clamped => uniform control flow)
    int ti = grp * WAVES + wave;
    if (ti > NTILES - 1) ti = NTILES - 1;
    const int t0 = ti * TILE_P;

    const int rloc = lane & 15;          // row within tile (A layout, both halves)
    const int koff = (lane >> 4) << 1;   // 0 for lanes 0-15, 2 for lanes 16-31
    int rowt = t0 + rloc;
    if (rowt > T_ - 1) rowt = T_ - 1;    // clamp (only affects padded tail tile)
    const float* xr  = x + (((size_t)bb) * T_ + rowt) * C_;
    const float* bt0 = &Bt[(lane & 15) * BT_PITCH];
    const float* bt1 = &Bt[((lane & 15) + 16) * BT_PITCH];

    v8f acc0 = {0.f,0.f,0.f,0.f,0.f,0.f,0.f,0.f};
    v8f acc1 = {0.f,0.f,0.f,0.f,0.f,0.f,0.f,0.f};
    float sq = 0.0f;                     // partial |x row|^2 over this lane's K slots

    // ---- GEMM: G(16x24) = Xtile(16x128) * Bt(128x24), f32 WMMA, 32 k-steps
    #pragma unroll 4
    for (int kb = 0; kb < C_; kb += 4) {
        v2f a  = *(const v2f*)(xr  + kb + koff);
        v2f b0 = *(const v2f*)(bt0 + kb + koff);
        v2f b1 = *(const v2f*)(bt1 + kb + koff);
        acc0 = __builtin_amdgcn_wmma_f32_16x16x4_f32(false, a, false, b0,
                                                     (short)0, acc0, false, false);
        acc1 = __builtin_amdgcn_wmma_f32_16x16x4_f32(false, a, false, b1,
                                                     (short)0, acc1, false, false);
        sq = fmaf(a.x, a.x, sq);
        sq = fmaf(a.y, a.y, sq);
    }

    // row norms: lane L covers cols {0,1 mod 4}, lane L+16 covers {2,3 mod 4}
    float sqo = __shfl_xor(sq, 16, 32);
    if (lane < 16) xnl[wave][lane] = sq + sqo;

    // dump G tile: C/D layout -> VGPR v holds M = v + 8*(lane>=16), N = lane%16
    const int n0  = lane & 15;
    const int mhi = (lane >> 4) << 3;
    #pragma unroll
    for (int v = 0; v < 8; ++v) {
        const int r = v + mhi;
        Gl[wave][r * NW_ + n0] = acc0[v];
        if (n0 + 16 < NW_) Gl[wave][r * NW_ + n0 + 16] = acc1[v];
    }
    __syncthreads();   // uniform: makes per-wave LDS tiles visible

    // ---- DTW + traceback: 14 windows x 8 filters = 112 units per wave
    for (int iter = 0; iter < 4; ++iter) {
        const int u = iter * 32 + lane;
        if (u >= TILE_P * F_) break;
        const int pl = u >> 3;
        const int f  = u & 7;
        const int p  = t0 + pl;
        if (p >= P_) continue;

        float xns[3] = { xnl[wave][pl], xnl[wave][pl + 1], xnl[wave][pl + 2] };
        float dots[3][3], D[3][3];
        #pragma unroll
        for (int i = 0; i < 3; ++i) {
            #pragma unroll
            for (int j = 0; j < 3; ++j) {
                const float d = Gl[wave][(pl + i) * NW_ + j * 8 + f];
                dots[i][j] = d;
                const float s = xns[i] + wn[j * 8 + f] - 2.0f * d;
                D[i][j] = sqrtf(fmaxf(s, 0.0f));
            }
        }

        // DTW cost matrix (4x4 with BIG borders)
        float cost[4][4];
        cost[0][0] = 0.0f; cost[0][1] = BIGV; cost[0][2] = BIGV; cost[0][3] = BIGV;
        cost[1][0] = BIGV; cost[2][0] = BIGV; cost[3][0] = BIGV;
        #pragma unroll
        for (int i = 1; i <= 3; ++i) {
            #pragma unroll
            for (int j = 1; j <= 3; ++j) {
                const float m = fminf(fminf(cost[i][j - 1], cost[i - 1][j - 1]),
                                      cost[i - 1][j]);
                cost[i][j] = D[i - 1][j - 1] + m;
            }
        }

        // traceback: argmin over [diag, left, up], first-min tie-break like jnp.argmin
        float acc = dots[2][2];
        int i = 3, j = 3;
        for (int s5 = 0; s5 < 5; ++s5) {
            const float cd = cost[i - 1][j - 1];
            const float cl = cost[i][j - 1];
            const float cu = cost[i - 1][j];
            const int m = (cd <= cl && cd <= cu) ? 0 : ((cl <= cu) ? 1 : 2);
            const int i2 = i - ((m == 1) ? 0 : 1);
            const int j2 = j - ((m == 2) ? 0 : 1);
            const bool nxt = (i2 > 0) && (j2 > 0);
            if (nxt) acc += dots[i2 - 1][j2 - 1];
            i = i2; j = j2;
            if (!nxt) break;
        }

        out[(((size_t)bb) * P_ + p) * F_ + f] = fmaxf(acc + bias[f], 0.0f);
    }
}

extern "C" void kernel_launch(void* const* d_in, const int* in_sizes, int n_in,
                              void* d_out, int out_size, void* d_ws, size_t ws_size,
                              hipStream_t stream) {
    (void)in_sizes; (void)n_in; (void)d_ws; (void)ws_size; (void)out_size;
    const float* x    = (const float*)d_in[0];
    const float* w    = (const float*)d_in[1];
    const float* bias = (const float*)d_in[2];
    float* out        = (float*)d_out;

    dim3 grid(B_ * GROUPS);   // 32 * 19 = 608 blocks
    dim3 block(256);          // 8 waves of 32
    dwa_cnn_fused<<<grid, block, 0, stream>>>(x, w, bias, out);
}